// ContrastiveWrapper_31782757991084
// MI455X (gfx1250) — compile-verified
//
#include <hip/hip_runtime.h>
#include <cstdint>
#include <math.h>

typedef __attribute__((ext_vector_type(2))) float v2f;
typedef __attribute__((ext_vector_type(8))) float v8f;

#define NROWS   8192
#define DIN     512
#define HIDDIM  1024
#define EMBDIM  64
#define NCON    62
#define MARGINF 0.01f

// ---------------------------------------------------------------------------
// Fused GEMM:  C[M,N] = act(A[M,K] @ B[K,N] + bias[N])   (all fp32)
// f32 WMMA 16x16x4, register-blocked: one wave computes a 16x64 strip of C
// (4 N-tiles), so each A fragment load feeds 4 WMMAs. K-loop unrolled x4 to
// let the scheduler pipeline 16 WMMAs against batched global loads.
// blockDim = (32, WY); block covers 16 rows x (64*WY) cols.
// K must be a multiple of 4; N a multiple of 64*WY*gridDim.y coverage.
//
// f32 operand layout (CDNA5 ISA):
//   A 16x4 : lanes 0-15 rows M=0..15, v.x=K0,v.y=K1 ; lanes 16-31 v.x=K2,v.y=K3
//   B 4x16 : lanes 0-15 cols N=0..15, v.x=K0,v.y=K1 ; lanes 16-31 v.x=K2,v.y=K3
//   C/D    : VGPR r -> M=r (lanes 0-15) / M=r+8 (lanes 16-31), N=lane&15
// ---------------------------------------------------------------------------
template <bool TANH>
__global__ void wmma_gemm_f32(const float* __restrict__ A, int lda,
                              const float* __restrict__ B, int ldb,
                              const float* __restrict__ bias,
                              float* __restrict__ C, int ldc, int K) {
  const int lane   = threadIdx.x;          // 0..31
  const int laneHi = lane >> 4;            // 0 | 1
  const int lane15 = lane & 15;
  const int mTile  = blockIdx.x;
  const int nGroup = blockIdx.y * blockDim.y + threadIdx.y;  // 64-wide group
  const int colBase = nGroup * 64;

  const int rowA = mTile * 16 + lane15;

  const float* Ap = A + (long)rowA * lda + laneHi * 2;        // K = k+laneHi*2
  const float* Bp = B + (long)(laneHi * 2) * ldb + colBase + lane15;

  v8f c0 = {}, c1 = {}, c2 = {}, c3 = {};
#pragma unroll 4
  for (int k = 0; k < K; k += 4) {
    v2f a;
    a.x = Ap[k];
    a.y = Ap[k + 1];
    const float* Bk = Bp + (long)k * ldb;
    v2f b0, b1, b2, b3;
    b0.x = Bk[0];   b0.y = Bk[ldb + 0];
    b1.x = Bk[16];  b1.y = Bk[ldb + 16];
    b2.x = Bk[32];  b2.y = Bk[ldb + 32];
    b3.x = Bk[48];  b3.y = Bk[ldb + 48];
    c0 = __builtin_amdgcn_wmma_f32_16x16x4_f32(false, a, false, b0, (short)0, c0, false, false);
    c1 = __builtin_amdgcn_wmma_f32_16x16x4_f32(false, a, false, b1, (short)0, c1, false, false);
    c2 = __builtin_amdgcn_wmma_f32_16x16x4_f32(false, a, false, b2, (short)0, c2, false, false);
    c3 = __builtin_amdgcn_wmma_f32_16x16x4_f32(false, a, false, b3, (short)0, c3, false, false);
  }

  float* Crow = C + (long)(mTile * 16 + laneHi * 8) * ldc + colBase + lane15;
  v8f accs[4] = {c0, c1, c2, c3};
#pragma unroll
  for (int t = 0; t < 4; ++t) {
    const float bn = bias[colBase + t * 16 + lane15];
#pragma unroll
    for (int r = 0; r < 8; ++r) {
      float v = accs[t][r] + bn;
      if (TANH) v = tanhf(v);
      Crow[(long)r * ldc + t * 16] = v;
    }
  }
}

// ---------------------------------------------------------------------------
// Prep: conserved (cols 2..63 of emb) into zero-padded 64-wide buffer,
// per-row sum of squares, labels (col 0 of x) as int.
// ---------------------------------------------------------------------------
__global__ void prep_pairwise(const float* __restrict__ x,    // 8192 x 513
                              const float* __restrict__ emb,  // 8192 x 64
                              float* __restrict__ cp,         // 8192 x 64
                              float* __restrict__ sq,         // 8192
                              int* __restrict__ labels) {     // 8192
  int row = blockIdx.x * blockDim.x + threadIdx.x;
  if (row >= NROWS) return;
  float s = 0.f;
#pragma unroll 8
  for (int c = 0; c < NCON; ++c) {
    float v = emb[row * EMBDIM + 2 + c];
    cp[row * EMBDIM + c] = v;
    s += v * v;
  }
  cp[row * EMBDIM + 62] = 0.f;
  cp[row * EMBDIM + 63] = 0.f;
  sq[row] = s;
  labels[row] = (int)x[(long)row * (DIN + 1)];
}

// ---------------------------------------------------------------------------
// Pairwise contrastive reduction. Each wave computes one 16x16 tile of the
// gram matrix conserved @ conserved^T with 16 f32 WMMAs (K=64, zero padded),
// converts to D_mean in-register, accumulates hinge terms, then block-level
// double reduction + 3 f64 atomics. Never materializes the 8192^2 matrix.
// blockDim=(32,8), grid=(512, 64): block covers rows [bx*16,+16), cols 8*16.
// ---------------------------------------------------------------------------
__global__ void pairwise_loss(const float* __restrict__ cp,
                              const float* __restrict__ sq,
                              const int* __restrict__ labels,
                              double* __restrict__ acc) {
  const int lane   = threadIdx.x;
  const int wave   = threadIdx.y;
  const int laneHi = lane >> 4;
  const int lane15 = lane & 15;
  const int i0 = blockIdx.x * 16;
  const int j0 = (blockIdx.y * blockDim.y + wave) * 16;

  // A = cp rows i (16x64); B = cp^T (64x16): B[k][j] = cp[j*64+k].
  const float2* Ap = (const float2*)(cp + (long)(i0 + lane15) * EMBDIM + laneHi * 2);
  const float2* Bp = (const float2*)(cp + (long)(j0 + lane15) * EMBDIM + laneHi * 2);

  v8f g = {};
#pragma unroll
  for (int k4 = 0; k4 < EMBDIM / 4; ++k4) {
    float2 af = Ap[k4 * 2];  // floats k4*4 + laneHi*2 .. +1
    float2 bf = Bp[k4 * 2];
    v2f a; a.x = af.x; a.y = af.y;
    v2f b; b.x = bf.x; b.y = bf.y;
    g = __builtin_amdgcn_wmma_f32_16x16x4_f32(false, a, false, b,
                                              (short)0, g, false, false);
  }

  const int   j   = j0 + lane15;
  const float sqj = sq[j];
  const int   lj  = labels[j];

  float s_diff = 0.f, s_same = 0.f;
  int   n_diff = 0;
#pragma unroll
  for (int r = 0; r < 8; ++r) {
    int   i = i0 + r + laneHi * 8;
    float d = fmaxf(sq[i] + sqj - 2.0f * g[r], 0.0f) * (1.0f / (float)NCON);
    if (labels[i] != lj) {
      s_diff += d;
      n_diff += 1;
    } else {
      s_same += fmaxf(0.0f, MARGINF - d);
    }
  }

  __shared__ double red[3][256];
  const int tid = wave * 32 + lane;
  red[0][tid] = (double)s_diff;
  red[1][tid] = (double)s_same;
  red[2][tid] = (double)n_diff;
  __syncthreads();
  for (int s = 128; s > 0; s >>= 1) {
    if (tid < s) {
      red[0][tid] += red[0][tid + s];
      red[1][tid] += red[1][tid + s];
      red[2][tid] += red[2][tid + s];
    }
    __syncthreads();
  }
  if (tid == 0) {
    atomicAdd(&acc[0], red[0][0]);
    atomicAdd(&acc[1], red[1][0]);
    atomicAdd(&acc[2], red[2][0]);
  }
}

__global__ void zero_acc(double* __restrict__ acc) {
  if (threadIdx.x < 3) acc[threadIdx.x] = 0.0;
}

__global__ void finalize_loss(const double* __restrict__ acc,
                              float* __restrict__ out2) {
  double s_diff = acc[0], s_same = acc[1], nd = acc[2];
  out2[0] = (float)(s_diff / (nd + 1.0));
  out2[1] = (float)(s_same / ((double)NROWS * (double)NROWS - nd + 1.0));
}

// ---------------------------------------------------------------------------
extern "C" void kernel_launch(void* const* d_in, const int* in_sizes, int n_in,
                              void* d_out, int out_size, void* d_ws, size_t ws_size,
                              hipStream_t stream) {
  const float* x  = (const float*)d_in[0];  // 8192 x 513 (col 0 = labels)
  const float* W1 = (const float*)d_in[1];  // 512 x 1024
  const float* b1 = (const float*)d_in[2];
  const float* W2 = (const float*)d_in[3];  // 1024 x 64
  const float* b2 = (const float*)d_in[4];
  const float* W3 = (const float*)d_in[5];  // 64 x 1024
  const float* b3 = (const float*)d_in[6];
  const float* W4 = (const float*)d_in[7];  // 1024 x 512
  const float* b4 = (const float*)d_in[8];
  float* out = (float*)d_out;               // decoded (8192*512) ++ [C_sim, C_diff]

  // Workspace carve-up (~38 MB): H (shared by h1/h2), emb, padded conserved,
  // sq, labels, double accumulators.
  float* H   = (float*)d_ws;                               // 8192*1024
  float* Emb = H + (size_t)NROWS * HIDDIM;                 // 8192*64
  float* Cp  = Emb + (size_t)NROWS * EMBDIM;               // 8192*64
  float* Sq  = Cp + (size_t)NROWS * EMBDIM;                // 8192
  int*   Lbl = (int*)(Sq + NROWS);                         // 8192
  double* Acc = (double*)(((uintptr_t)(Lbl + NROWS) + 15) & ~(uintptr_t)15);

  zero_acc<<<1, 32, 0, stream>>>(Acc);

  // h1 = tanh(data @ W1 + b1)    data = x[:,1:], lda = 513 (row-major slice)
  // N=1024 -> 16 groups of 64; 4 waves/block -> grid.y = 4
  wmma_gemm_f32<true><<<dim3(NROWS / 16, 4), dim3(32, 4), 0, stream>>>(
      x + 1, DIN + 1, W1, HIDDIM, b1, H, HIDDIM, DIN);

  // emb = h1 @ W2 + b2           N=64 -> 1 group; 1 wave/block
  wmma_gemm_f32<false><<<dim3(NROWS / 16, 1), dim3(32, 1), 0, stream>>>(
      H, HIDDIM, W2, EMBDIM, b2, Emb, EMBDIM, HIDDIM);

  // conserved slice + norms + labels
  prep_pairwise<<<NROWS / 256, 256, 0, stream>>>(x, Emb, Cp, Sq, Lbl);

  // h2 = tanh(emb @ W3 + b3)     (reuse H)  N=1024 -> grid.y = 4
  wmma_gemm_f32<true><<<dim3(NROWS / 16, 4), dim3(32, 4), 0, stream>>>(
      Emb, EMBDIM, W3, HIDDIM, b3, H, HIDDIM, EMBDIM);

  // decoded = h2 @ W4 + b4 -> d_out   N=512 -> 8 groups; grid.y = 2
  wmma_gemm_f32<false><<<dim3(NROWS / 16, 2), dim3(32, 4), 0, stream>>>(
      H, HIDDIM, W4, DIN, b4, out, DIN, HIDDIM);

  // Contrastive reduction over all 8192^2 pairs (tile-fused, no gram in HBM).
  pairwise_loss<<<dim3(NROWS / 16, NROWS / (16 * 8)), dim3(32, 8), 0, stream>>>(
      Cp, Sq, Lbl, Acc);

  finalize_loss<<<1, 1, 0, stream>>>(Acc, out + (size_t)NROWS * DIN);
}